// MaskToken_22428319220331
// MI455X (gfx1250) — compile-verified
//
#include <hip/hip_runtime.h>
#include <cstdint>
#include <cstddef>

// Problem constants (match reference)
#define B_    4
#define L_    4096
#define D_    1024
#define NKEEP 1024
#define NDROP 3072

// ---- gfx1250 async global<->LDS DMA path (probe-guarded) --------------------
#if defined(__has_builtin)
#  if __has_builtin(__builtin_amdgcn_global_load_async_to_lds_b128) && \
      __has_builtin(__builtin_amdgcn_global_store_async_from_lds_b128) && \
      __has_builtin(__builtin_amdgcn_s_wait_asynccnt)
#    define HAS_ASYNC_LDS 1
#  endif
#endif
#ifndef HAS_ASYNC_LDS
#  define HAS_ASYNC_LDS 0
#endif

// Builtin signature (from hipcc diagnostic): params are int4* in AS(1)/AS(3).
typedef __attribute__((ext_vector_type(4))) int v4i;
typedef __attribute__((address_space(1))) v4i as1_v4i;  // global (64-bit ptr)
typedef __attribute__((address_space(3))) v4i as3_v4i;  // LDS (32-bit ptr)

__device__ __forceinline__ as1_v4i* as_global_v4(const void* p) {
  // numeric value of a generic global pointer == AS1 address
  return (as1_v4i*)(unsigned long long)p;
}
__device__ __forceinline__ as3_v4i* as_lds_v4(const void* p) {
  // low 32 bits of a generic LDS pointer == AS3 address
  return (as3_v4i*)(unsigned)(unsigned long long)p;
}

// Copy one 4KB row (D_=1024 floats) with 256 threads, 16B per lane.
// Async path: HBM -> LDS -> HBM, no VGPR data round trip, tracked by ASYNCcnt.
// Each lane touches only its own 16B LDS slot, so a per-wave s_wait_asynccnt
// between the load and the store is sufficient (no block barrier needed).
__device__ __forceinline__ void row_copy_4k(const float* __restrict__ src,
                                            float* __restrict__ dst) {
  const int t = threadIdx.x;  // 0..255
#if HAS_ASYNC_LDS
  __shared__ __align__(16) float tile[D_];
  const float* sp = src + t * 4;
  float*       lp = tile + t * 4;
  float*       dp = dst + t * 4;
  __builtin_amdgcn_global_load_async_to_lds_b128(as_global_v4(sp), as_lds_v4(lp), 0, 0);
  __builtin_amdgcn_s_wait_asynccnt(0);
  __builtin_amdgcn_global_store_async_from_lds_b128(as_global_v4(dp), as_lds_v4(lp), 0, 0);
  __builtin_amdgcn_s_wait_asynccnt(0);
#else
  const float4* s4 = reinterpret_cast<const float4*>(src) + t;
  float4*       d4 = reinterpret_cast<float4*>(dst) + t;
  *d4 = *s4;
#endif
}

// outputs_dropped[b,k,:] = inputs[b, idx_keep[k], :]
// grid = B_*NKEEP blocks, 256 threads.
__global__ void mt_gather_rows(const float* __restrict__ in,
                               const int*   __restrict__ idx,
                               float*       __restrict__ out) {
  const int row = blockIdx.x;          // b*NKEEP + k
  const int b   = row >> 10;           // NKEEP == 1024
  const int k   = row & (NKEEP - 1);
  const int sl  = idx[k];              // blockIdx-uniform -> scalar load
  row_copy_4k(in + ((size_t)b * L_ + (size_t)sl) * D_,
              out + (size_t)row * D_);
}

// outputs_masked[b,l,:] = kept(l) ? inputs[b,l,:] : mask_embedding[:]
// grid = B_*L_ blocks (blockIdx.x == flat row b*L+l), 256 threads.
// Keep/drop via branchless binary search over sorted idx_keep (all operands
// blockIdx-derived -> stays in SGPRs / s_load).
__global__ void mt_masked_rows(const float* __restrict__ in,
                               const float* __restrict__ me,
                               const int*   __restrict__ idx,
                               float*       __restrict__ out) {
  const int row = blockIdx.x;
  const int l   = row & (L_ - 1);      // L_ == 4096
  int lo = 0;
#pragma unroll
  for (int step = NKEEP >> 1; step > 0; step >>= 1) {
    if (idx[lo + step - 1] < l) lo += step;
  }
  const bool kept = (idx[lo] == l);
  const float* src = kept ? (in + (size_t)row * D_) : me;  // me is L2-hot (4KB)
  row_copy_4k(src, out + (size_t)row * D_);
}

// mask_drop[l] = kept(l) ? 0 : 1 ; idx_out[k] = (float)idx_keep[k]
// grid = L_/256 blocks, 256 threads. Per-thread binary search (VALU).
__global__ void mt_small(const int* __restrict__ idx,
                         float*     __restrict__ mask_drop,
                         float*     __restrict__ idx_out) {
  const int l = blockIdx.x * blockDim.x + threadIdx.x;  // 0..L_-1
  int lo = 0;
#pragma unroll
  for (int step = NKEEP >> 1; step > 0; step >>= 1) {
    if (idx[lo + step - 1] < l) lo += step;
  }
  const bool kept = (idx[lo] == l);
  mask_drop[l] = kept ? 0.0f : 1.0f;
  if (l < NKEEP) idx_out[l] = (float)idx[l];
}

extern "C" void kernel_launch(void* const* d_in, const int* in_sizes, int n_in,
                              void* d_out, int out_size, void* d_ws, size_t ws_size,
                              hipStream_t stream) {
  (void)in_sizes; (void)n_in; (void)out_size; (void)d_ws; (void)ws_size;

  const float* inputs = (const float*)d_in[0];   // [B, L, D] f32
  const float* me     = (const float*)d_in[1];   // [D] f32
  const int*   idx    = (const int*)  d_in[2];   // [NKEEP] i32 (sorted)

  float* out          = (float*)d_out;
  float* out_dropped  = out;                                       // B*NKEEP*D
  float* out_masked   = out_dropped + (size_t)B_ * NKEEP * D_;     // B*L*D
  float* out_maskdrop = out_masked  + (size_t)B_ * L_ * D_;        // L
  float* out_idx      = out_maskdrop + L_;                         // NKEEP

  mt_gather_rows<<<B_ * NKEEP, 256, 0, stream>>>(inputs, idx, out_dropped);
  mt_masked_rows<<<B_ * L_,    256, 0, stream>>>(inputs, me, idx, out_masked);
  mt_small      <<<L_ / 256,   256, 0, stream>>>(idx, out_maskdrop, out_idx);
}